// GraphEncoder_15547781611788
// MI455X (gfx1250) — compile-verified
//
#include <hip/hip_runtime.h>

#define N_NODES 81920
#define N_EDGES 1310720
#define NODES_PER_GRAPH 320
#define N_GRAPHS 256
#define DIM 64

typedef float v2f __attribute__((ext_vector_type(2)));
typedef float v8f __attribute__((ext_vector_type(8)));

__device__ __forceinline__ void atomic_add_f32(float* p, float v) {
    // lowers to global_atomic_add_f32 (no-return) on gfx1250
    (void)__hip_atomic_fetch_add(p, v, __ATOMIC_RELAXED, __HIP_MEMORY_SCOPE_AGENT);
}

__global__ void k_fill_zero(float4* __restrict__ p, int n4) {
    int i = blockIdx.x * blockDim.x + threadIdx.x;
    if (i < n4) p[i] = make_float4(0.f, 0.f, 0.f, 0.f);
}

__global__ void k_count(const int* __restrict__ dst, float* __restrict__ cnt) {
    int e = blockIdx.x * blockDim.x + threadIdx.x;
    if (e < N_EDGES) atomic_add_f32(&cnt[dst[e]], 1.0f);
}

__global__ void k_finalize_inv(float* __restrict__ cnt) {
    int i = blockIdx.x * blockDim.x + threadIdx.x;
    if (i < N_NODES) cnt[i] = 1.0f / fmaxf(cnt[i], 1.0f);
}

// One wave (32 lanes) per edge; each lane moves 2 floats (float2) -> 256B
// coalesced gather + coalesced f32 atomics into the L2-resident agg buffer.
__global__ void k_scatter(const float* __restrict__ h, const int* __restrict__ src,
                          const int* __restrict__ dst, float* __restrict__ agg) {
    unsigned gid = blockIdx.x * blockDim.x + threadIdx.x;
    unsigned e = gid >> 5;
    if (e >= N_EDGES) return;
    int d = (gid & 31u) * 2;
    int s = src[e];
    int t = dst[e];
    const float2 v = *(const float2*)(h + (size_t)s * DIM + d);
    float* out = agg + (size_t)t * DIM + d;
    atomic_add_f32(out, v.x);
    atomic_add_f32(out + 1, v.y);
}

// Fused SAGE layer GEMM: out = relu?( (agg * inv_deg) @ Wl + bl + h @ Wr )
// One wave per 16-row tile; 4 column tiles of 16; K=64 as 16 steps of
// v_wmma_f32_16x16x4_f32 per (tile, matrix). Mean-divide fused into the A
// fragment, bias into accumulator init, ReLU into the epilogue.
// NOTE: `agg` and `out` may alias (in-place): each wave reads only its own
// 16 rows in the K loop before its stores, so this is race-free.
__global__ void k_sage_gemm(const float* agg, const float* __restrict__ inv,
                            const float* __restrict__ h,
                            const float* __restrict__ Wl, const float* __restrict__ bl,
                            const float* __restrict__ Wr,
                            float* out, int relu) {
    int wave = blockIdx.x * (blockDim.x >> 5) + (threadIdx.x >> 5);
    int lane = threadIdx.x & 31;
    int row0 = wave * 16;
    if (row0 >= N_NODES) return;
    int m  = lane & 15;   // M index within tile (and N index for B/C cols)
    int kh = lane >> 4;   // lane half selects K pair {0,1} vs {2,3}

    int row = row0 + m;
    float invd = inv[row];
    const float* aggRow = agg + (size_t)row * DIM;
    const float* hRow   = h   + (size_t)row * DIM;

    v8f acc[4];
#pragma unroll
    for (int t = 0; t < 4; ++t) {
        float b = bl[t * 16 + m];
#pragma unroll
        for (int r = 0; r < 8; ++r) acc[t][r] = b;
    }

#pragma unroll
    for (int k0 = 0; k0 < DIM; k0 += 4) {
        // A fragments (16x4 f32): VGPR0/1 hold K = k0+2*kh, k0+2*kh+1
        const float2 ta = *(const float2*)(aggRow + k0 + 2 * kh);
        const float2 th = *(const float2*)(hRow   + k0 + 2 * kh);
        v2f aA, aH;
        aA.x = ta.x * invd; aA.y = ta.y * invd;
        aH.x = th.x;        aH.y = th.y;
#pragma unroll
        for (int t = 0; t < 4; ++t) {
            int col = t * 16 + m;
            // B fragments (4x16 f32): rows K = k0+2*kh, k0+2*kh+1; col = lane&15
            v2f bL, bR;
            bL.x = Wl[(k0 + 2 * kh)     * DIM + col];
            bL.y = Wl[(k0 + 2 * kh + 1) * DIM + col];
            bR.x = Wr[(k0 + 2 * kh)     * DIM + col];
            bR.y = Wr[(k0 + 2 * kh + 1) * DIM + col];
            acc[t] = __builtin_amdgcn_wmma_f32_16x16x4_f32(
                false, aA, false, bL, (short)0, acc[t], false, false);
            acc[t] = __builtin_amdgcn_wmma_f32_16x16x4_f32(
                false, aH, false, bR, (short)0, acc[t], false, false);
        }
    }

    // C/D layout: VGPR r -> row (r + 8*kh), col = t*16 + (lane&15)
#pragma unroll
    for (int t = 0; t < 4; ++t) {
#pragma unroll
        for (int r = 0; r < 8; ++r) {
            float v = acc[t][r];
            if (relu) v = fmaxf(v, 0.0f);
            int orow = row0 + r + kh * 8;
            out[(size_t)orow * DIM + t * 16 + m] = v;
        }
    }
}

__global__ void k_gather_out(const float* __restrict__ h, float* __restrict__ out) {
    int i = blockIdx.x * blockDim.x + threadIdx.x;
    if (i < N_GRAPHS * DIM) {
        int g = i / DIM;
        int d = i - g * DIM;
        out[i] = h[(size_t)g * NODES_PER_GRAPH * DIM + d];
    }
}

extern "C" void kernel_launch(void* const* d_in, const int* in_sizes, int n_in,
                              void* d_out, int out_size, void* d_ws, size_t ws_size,
                              hipStream_t stream) {
    (void)in_sizes; (void)n_in; (void)out_size; (void)ws_size;
    const float* x    = (const float*)d_in[0];
    const int*   ei   = (const int*)d_in[1];
    // d_in[2] = batch (unused: first node of graph g is g*NODES_PER_GRAPH)
    const float* Wl0 = (const float*)d_in[3];
    const float* bl0 = (const float*)d_in[4];
    const float* Wr0 = (const float*)d_in[5];
    const float* Wl1 = (const float*)d_in[6];
    const float* bl1 = (const float*)d_in[7];
    const float* Wr1 = (const float*)d_in[8];
    const float* Wl2 = (const float*)d_in[9];
    const float* bl2 = (const float*)d_in[10];
    const float* Wr2 = (const float*)d_in[11];

    const int* src = ei;
    const int* dst = ei + N_EDGES;

    float* ws0 = (float*)d_ws;                          // 81920*64 f32
    float* ws1 = ws0 + (size_t)N_NODES * DIM;           // 81920*64 f32
    float* inv = ws1 + (size_t)N_NODES * DIM;           // 81920 f32

    const int NF   = N_NODES * DIM;
    const int NF4  = NF / 4;
    const int TPB  = 256;
    const int scatterBlocks = (N_EDGES * 32) / TPB;     // exact
    const int gemmBlocks    = (N_NODES / 16) / 8;       // 8 waves/block, exact

    // Degrees -> inverse degree (shared by all three layers)
    k_fill_zero<<<(N_NODES / 4 + TPB - 1) / TPB, TPB, 0, stream>>>((float4*)inv, N_NODES / 4);
    k_count<<<(N_EDGES + TPB - 1) / TPB, TPB, 0, stream>>>(dst, inv);
    k_finalize_inv<<<(N_NODES + TPB - 1) / TPB, TPB, 0, stream>>>(inv);

    // Layer 0: h = x, agg = ws0, out = ws1 (relu)
    k_fill_zero<<<(NF4 + TPB - 1) / TPB, TPB, 0, stream>>>((float4*)ws0, NF4);
    k_scatter<<<scatterBlocks, TPB, 0, stream>>>(x, src, dst, ws0);
    k_sage_gemm<<<gemmBlocks, TPB, 0, stream>>>(ws0, inv, x, Wl0, bl0, Wr0, ws1, 1);

    // Layer 1: h = ws1, agg = ws0, out = ws0 in-place (relu)
    k_fill_zero<<<(NF4 + TPB - 1) / TPB, TPB, 0, stream>>>((float4*)ws0, NF4);
    k_scatter<<<scatterBlocks, TPB, 0, stream>>>(ws1, src, dst, ws0);
    k_sage_gemm<<<gemmBlocks, TPB, 0, stream>>>(ws0, inv, ws1, Wl1, bl1, Wr1, ws0, 1);

    // Layer 2: h = ws0, agg = ws1, out = ws1 in-place (no relu)
    k_fill_zero<<<(NF4 + TPB - 1) / TPB, TPB, 0, stream>>>((float4*)ws1, NF4);
    k_scatter<<<scatterBlocks, TPB, 0, stream>>>(ws0, src, dst, ws1);
    k_sage_gemm<<<gemmBlocks, TPB, 0, stream>>>(ws1, inv, ws0, Wl2, bl2, Wr2, ws1, 0);

    // Gather node 0 of each graph
    k_gather_out<<<(N_GRAPHS * DIM + TPB - 1) / TPB, TPB, 0, stream>>>(ws1, (float*)d_out);
}